// HardBinaryConv_72928544686678
// MI455X (gfx1250) — compile-verified
//
#include <hip/hip_runtime.h>

// ---------------------------------------------------------------------------
// HardBinaryConv (binarized 3x3 conv, chunked saturating GEMM) for gfx1250.
//
//   psum[n,b,l,c] = sum_{t<64} A[b,l,n*64+t] * sign(W)[n*64+t,c]
//   y[b,c,l]      = clip-chain_n(psum) * scale[c]
//
// Matrix core: V_WMMA_F32_16X16X4_F32 (exact fp32), 16 chained per chunk.
// psum writes (925 MB, the roofline term) are staged in LDS and drained as
// contiguous 2304-byte rows with float4 stores.
// ---------------------------------------------------------------------------

typedef __attribute__((ext_vector_type(2))) float v2f;
typedef __attribute__((ext_vector_type(8))) float v8f;

#define CIN   256
#define COUT  256
#define HW    56
#define LTOT  (HW * HW)        // 3136 output positions
#define KTOT  (CIN * 9)        // 2304 reduction length
#define TCH   64               // chunk size along K
#define NT    36               // KTOT / TCH
#define BATCH 8
#define QMAXF 127.0f
#define QMINF (-128.0f)

#define WAVES    2             // waves (16-row L tiles) per workgroup
#define ASTRIDE  68            // 64 + 4: breaks stride-64 bank aliasing
#define PSTRIDE  580           // 16*36 + 4: keeps rows 16B-aligned (580*4=2320)

// ---------------- prep: binarize weights -> Wb[k][c], scale[c] -------------
__global__ __launch_bounds__(256) void hbc_prep_kernel(
    const float* __restrict__ weights,  // [COUT*KTOT] row-major per cout
    float* __restrict__ wb,             // out: [KTOT][COUT] (GEMM B layout)
    float* __restrict__ scale)          // out: [COUT]
{
    const int c = blockIdx.x;
    const int t = threadIdx.x;
    float s = 0.0f;
    for (int k = t; k < KTOT; k += 256) {
        float w = weights[(size_t)c * KTOT + k];
        s += fabsf(w);
        float sg = (w > 0.0f) ? 1.0f : ((w < 0.0f) ? -1.0f : 0.0f);
        wb[(size_t)k * COUT + c] = sg;
    }
    __shared__ float red[256];
    red[t] = s;
    __syncthreads();
    for (int off = 128; off > 0; off >>= 1) {
        if (t < off) red[t] += red[t + off];
        __syncthreads();
    }
    if (t == 0) scale[c] = red[0] * (1.0f / (float)KTOT);
}

// ---------------- main: chunked WMMA GEMM + saturating scan ----------------
__global__ __launch_bounds__(32 * WAVES) void hbc_satconv_kernel(
    const float* __restrict__ x,      // [B][CIN][56][56]
    const float* __restrict__ wb,     // [KTOT][COUT] binarized
    const float* __restrict__ scale,  // [COUT]
    float* __restrict__ y,            // [B][COUT][56][56]
    float* __restrict__ psum)         // [B][LTOT][COUT][NT]
{
    __shared__ float Blds[TCH][16];                       // 4 KB, shared by WG
    __shared__ float Alds[WAVES][16][ASTRIDE];            // 8.7 KB
    __shared__ __align__(16) float Plds[WAVES][16][PSTRIDE]; // 74.2 KB stage

    const int tid  = threadIdx.x;
    const int wave = tid >> 5;
    const int lane = tid & 31;
    const int half = lane >> 4;       // 0: lanes 0-15, 1: lanes 16-31
    const int l16  = lane & 15;

    const int b  = blockIdx.z;
    const int c0 = blockIdx.y * 16;                       // cout tile
    const int l0 = (blockIdx.x * WAVES + wave) * 16;      // L tile (per wave)

    const float* xb = x + (size_t)b * CIN * LTOT;

    v8f acc = {};                                         // saturating carry

    for (int n = 0; n < NT; ++n) {
        __syncthreads();  // previous chunk's WMMA reads done before overwrite

        // B tile: Blds[kk][cc] = Wb[n*64+kk][c0+cc]  (all threads, coalesced)
        for (int i = tid; i < TCH * 16; i += 32 * WAVES) {
            int kk = i >> 4, cc = i & 15;
            Blds[kk][cc] = wb[(size_t)(n * TCH + kk) * COUT + c0 + cc];
        }

        // A tile (per wave): on-the-fly im2col, zero padding at borders.
        for (int i = lane; i < 16 * TCH; i += 32) {
            int row = i >> 6, kk = i & 63;
            int k  = n * TCH + kk;
            int ci = k / 9;                 // div-by-const -> mulhi+shift
            int r9 = k - ci * 9;
            int kh = r9 / 3;
            int kw = r9 - kh * 3;
            int l  = l0 + row;
            int oh = l / HW;
            int ow = l - oh * HW;
            int ih = oh + kh - 1;
            int iw = ow + kw - 1;
            float v = 0.0f;
            if ((unsigned)ih < (unsigned)HW && (unsigned)iw < (unsigned)HW)
                v = xb[(ci * HW + ih) * HW + iw];
            Alds[wave][row][kk] = v;
        }
        __syncthreads();

        // 16 chained exact-f32 WMMAs: 16x16 tile, K advances by 4 each step.
        v8f tile = {};
#pragma unroll
        for (int kk = 0; kk < TCH; kk += 4) {
            v2f a, bf;
            // A 16x4 frag: lanes 0-15 {K=kk,kk+1}, lanes 16-31 {K=kk+2,kk+3}
            a.x  = Alds[wave][l16][kk + 2 * half];
            a.y  = Alds[wave][l16][kk + 2 * half + 1];
            // B 4x16 frag, mirrored K split across lane halves
            bf.x = Blds[kk + 2 * half][l16];
            bf.y = Blds[kk + 2 * half + 1][l16];
            tile = __builtin_amdgcn_wmma_f32_16x16x4_f32(
                false, a, false, bf, (short)0, tile, false, false);
        }

        // Stage chunk tile: element (M, N) -> Plds[wave][M][N*36 + n].
        // C/D layout: VGPR r holds M = r + 8*half, N = l16.
#pragma unroll
        for (int r = 0; r < 8; ++r) {
            int M = r + 8 * half;
            Plds[wave][M][l16 * NT + n] = tile[r];
        }

        // Saturating sequential accumulation (chunk 0 unclipped, per ref).
        if (n == 0) {
            acc = tile;
        } else {
#pragma unroll
            for (int r = 0; r < 8; ++r) {
                float s = acc[r] + tile[r];
                acc[r] = fminf(fmaxf(s, QMINF), QMAXF);
            }
        }
    }

    // ---- y: lane owns c = c0+l16, rows l = l0 + 8*half + r (contiguous) ----
    {
        float sc = scale[c0 + l16];
        float4 y0, y1;
        y0.x = acc[0] * sc; y0.y = acc[1] * sc;
        y0.z = acc[2] * sc; y0.w = acc[3] * sc;
        y1.x = acc[4] * sc; y1.y = acc[5] * sc;
        y1.z = acc[6] * sc; y1.w = acc[7] * sc;
        size_t yb = ((size_t)b * COUT + (size_t)(c0 + l16)) * LTOT
                    + (size_t)l0 + 8 * half;
        *(float4*)(y + yb)     = y0;
        *(float4*)(y + yb + 4) = y1;
    }

    __syncthreads();  // Plds fully populated

    // ---- psum drain: per output row, (c0..c0+15, n 0..35) is one contiguous
    // 576-float (2304 B) run -> fully coalesced float4 stores. -------------
    for (int t2 = lane; t2 < 16 * 144; t2 += 32) {
        int row = t2 / 144;                 // div-by-const
        int col = t2 - row * 144;           // float4 index within row
        float4 v = *(const float4*)&Plds[wave][row][col * 4];
        size_t gb = ((size_t)((size_t)b * LTOT + l0 + row) * COUT + c0) * NT;
        *(float4*)(psum + gb + (size_t)col * 4) = v;
    }
}

// ---------------------------------------------------------------------------
extern "C" void kernel_launch(void* const* d_in, const int* in_sizes, int n_in,
                              void* d_out, int out_size, void* d_ws, size_t ws_size,
                              hipStream_t stream) {
    const float* x       = (const float*)d_in[0];  // [8,256,56,56] fp32
    const float* weights = (const float*)d_in[1];  // [256*256*3*3,1] fp32

    float* y    = (float*)d_out;                                   // 6,422,528
    float* psum = (float*)d_out + (size_t)BATCH * COUT * LTOT;     // 231,211,008

    float* wb    = (float*)d_ws;                   // [KTOT*COUT] = 2.25 MB
    float* scale = wb + (size_t)KTOT * COUT;       // [COUT]

    hbc_prep_kernel<<<dim3(COUT), dim3(256), 0, stream>>>(weights, wb, scale);

    dim3 grid(LTOT / (16 * WAVES), COUT / 16, BATCH);  // (98, 16, 8)
    hbc_satconv_kernel<<<grid, dim3(32 * WAVES), 0, stream>>>(
        x, wb, scale, y, psum);
}